// BKTModel_64690797412665
// MI455X (gfx1250) — compile-verified
//
#include <hip/hip_runtime.h>
#include <math.h>

// Problem constants (match the reference)
#define B_   256
#define T_   50
#define NOBS 8192
#define NKC  50

typedef __attribute__((ext_vector_type(2))) float v2f;
typedef __attribute__((ext_vector_type(8))) float v8f;

// ---------------------------------------------------------------------------
// Kernel 1: recover assign[o] = argmax_k A[o,k]  (A is one-hot), and
// kc_probs[k*5+j] = sigmoid(kc_logits[k,j]).
// ---------------------------------------------------------------------------
__global__ void bkt_prep(const float* __restrict__ A,
                         const float* __restrict__ kc_logits,
                         int* __restrict__ assign,
                         float* __restrict__ kcp) {
    int gid = blockIdx.x * blockDim.x + threadIdx.x;
    if (gid < NOBS) {
        const float* row = A + (size_t)gid * NKC;
        int   best = 0;
        float bv   = row[0];
        #pragma unroll 5
        for (int k = 1; k < NKC; ++k) {
            float v = row[k];
            if (v > bv) { bv = v; best = k; }
        }
        assign[gid] = best;
    }
    if (gid < NKC * 5) {
        float x = kc_logits[gid];
        kcp[gid] = 1.0f / (1.0f + expf(-x));
    }
}

// ---------------------------------------------------------------------------
// Kernel 2: the sequential BKT recurrence. One workgroup, one thread per
// batch, per-batch KC state lives in LDS (50 x 256 floats = 51.2 KB of the
// 320 KB WGP LDS). Each thread only touches its own LDS column -> no
// cross-thread sync needed after init.
// ---------------------------------------------------------------------------
__global__ __launch_bounds__(B_) void bkt_recur(
        const int*   __restrict__ prev_kc,
        const int*   __restrict__ curr_kc,
        const float* __restrict__ prev_corr,
        const int*   __restrict__ assign,
        const float* __restrict__ kcp_g,
        float* __restrict__ probs_out,     // (B, T) -> d_out[0 .. 12800)
        float* __restrict__ kcstate_out) { // (B, NKC) workspace
    __shared__ float kcp[256];             // 250 used
    __shared__ float st[NKC * B_];         // st[k*256 + b]
    const int b = threadIdx.x;

    if (b < NKC * 5) kcp[b] = kcp_g[b];
    __syncthreads();

    // state0[b,k] = sigmoid(kc_logits[k,4])
    #pragma unroll 5
    for (int k = 0; k < NKC; ++k) st[k * B_ + b] = kcp[k * 5 + 4];

    // t = 0 (no update, prediction from initial state)
    {
        int   ck0 = curr_kc[b * T_];
        int   kc  = assign[ck0];
        float cs  = kcp[kc * 5 + 4];
        probs_out[b * T_] = kcp[kc * 5 + 2] * (1.0f - cs) + kcp[kc * 5 + 3] * cs;
    }

    for (int t = 1; t < T_; ++t) {
        int   pk  = prev_kc[b * T_ + t];
        int   ck  = curr_kc[b * T_ + t];
        float pco = prev_corr[b * T_ + t];   // exactly 0.0 or 1.0
        int   kp  = assign[pk];

        float pl = kcp[kp * 5 + 0];
        float pf = kcp[kp * 5 + 1];
        float g  = kcp[kp * 5 + 2];
        float s  = kcp[kp * 5 + 3];

        // p_out = pch^pcorr * (1-pch)^(1-pcorr), pcorr in {0,1}
        float p0 = pco * g + (1.0f - pco) * (1.0f - g);
        float p1 = pco * s + (1.0f - pco) * (1.0f - s);

        float ss   = st[kp * B_ + b];
        float filt = p1 * ss / (p0 * (1.0f - ss) + p1 * ss);
        float pred = pl * (1.0f - filt) + (1.0f - pf) * filt;
        st[kp * B_ + b] = pred;              // M[pk]-masked update, collapsed

        int   kc = assign[ck];
        float cs = st[kc * B_ + b];          // reads pred if kc == kp (matches ref order)
        probs_out[b * T_ + t] = kcp[kc * 5 + 2] * (1.0f - cs) + kcp[kc * 5 + 3] * cs;
    }

    #pragma unroll 5
    for (int k = 0; k < NKC; ++k) kcstate_out[b * NKC + k] = st[k * B_ + b];
}

// ---------------------------------------------------------------------------
// Kernel 3: state expansion as a WMMA GEMM:
//   state(256 x 8192) = kc_state(256 x 50) @ A^T(50 x 8192)
// Exact (A is 0/1, f32 accumulate). One wave per 16x16 output tile,
// V_WMMA_F32_16X16X4_F32, K padded 50 -> 52 (13 steps).
// A-operand layout (32b, 16x4): vgpr v, lanes 0-15 -> K=v, lanes 16-31 -> K=v+2.
// B-operand layout (32b, 4x16): vgpr v, lanes 0-15 -> row K=v, lanes 16-31 -> K=v+2.
// D layout: vgpr r, lanes 0-15 -> M=r, lanes 16-31 -> M=r+8.
// ---------------------------------------------------------------------------
__global__ __launch_bounds__(32) void bkt_expand_wmma(
        const float* __restrict__ kcs,     // (B, NKC)
        const float* __restrict__ Amat,    // (NOBS, NKC)
        float* __restrict__ state_out) {   // (B, NOBS) -> d_out + 12800
    const int tileN = blockIdx.x & 511;    // 8192 / 16 = 512 obs tiles
    const int tileM = blockIdx.x >> 9;     // 256  / 16 = 16  batch tiles
    const int n0 = tileN * 16;
    const int m0 = tileM * 16;

    const int lane = threadIdx.x;
    const int half = lane >> 4;            // 0: lanes 0-15, 1: lanes 16-31
    const int lrow = lane & 15;

    v8f acc = {};
    const float* arow = kcs  + (size_t)(m0 + lrow) * NKC;   // A operand row (batch)
    const float* brow = Amat + (size_t)(n0 + lrow) * NKC;   // B operand "column" (obs)

    #pragma unroll
    for (int kk = 0; kk < 52; kk += 4) {
        int k0 = kk + (half ? 2 : 0);
        int k1 = k0 + 1;
        float a0 = (k0 < NKC) ? arow[k0] : 0.0f;
        float a1 = (k1 < NKC) ? arow[k1] : 0.0f;
        float b0 = (k0 < NKC) ? brow[k0] : 0.0f;
        float b1 = (k1 < NKC) ? brow[k1] : 0.0f;
        v2f av = {a0, a1};
        v2f bv = {b0, b1};
        acc = __builtin_amdgcn_wmma_f32_16x16x4_f32(
                  /*neg_a=*/false, av, /*neg_b=*/false, bv,
                  /*c_mod=*/(short)0, acc, /*reuse_a=*/false, /*reuse_b=*/false);
    }

    const int rbase = m0 + (half ? 8 : 0);
    #pragma unroll
    for (int r = 0; r < 8; ++r) {
        state_out[(size_t)(rbase + r) * NOBS + n0 + lrow] = acc[r];
    }
}

// ---------------------------------------------------------------------------
// Launch: inputs are [prev_kc, curr_kc, prev_corr, kc_logits, A].
// d_out = concat(probs(256x50), state(256x8192)) = 2,109,952 floats.
// Workspace layout: assign int[8192] @0, kcp float[250] @32768,
// kc_state float[256*50] @33792  (total ~85 KB).
// ---------------------------------------------------------------------------
extern "C" void kernel_launch(void* const* d_in, const int* in_sizes, int n_in,
                              void* d_out, int out_size, void* d_ws, size_t ws_size,
                              hipStream_t stream) {
    const int*   prev_kc   = (const int*)  d_in[0];
    const int*   curr_kc   = (const int*)  d_in[1];
    const float* prev_corr = (const float*)d_in[2];
    const float* kc_logits = (const float*)d_in[3];
    const float* Amat      = (const float*)d_in[4];

    float* out       = (float*)d_out;
    float* probs_out = out;                 // 256*50
    float* state_out = out + B_ * T_;       // 256*8192

    char*  ws       = (char*)d_ws;
    int*   assign   = (int*)  (ws + 0);
    float* kcp      = (float*)(ws + 32768);
    float* kcstate  = (float*)(ws + 33792);

    bkt_prep<<<dim3((NOBS + 255) / 256), dim3(256), 0, stream>>>(
        Amat, kc_logits, assign, kcp);

    bkt_recur<<<dim3(1), dim3(B_), 0, stream>>>(
        prev_kc, curr_kc, prev_corr, assign, kcp, probs_out, kcstate);

    bkt_expand_wmma<<<dim3((B_ / 16) * (NOBS / 16)), dim3(32), 0, stream>>>(
        kcstate, Amat, state_out);
}